// ChessformerAdapter_73976516706470
// MI455X (gfx1250) — compile-verified
//
#include <hip/hip_runtime.h>
#include <hip/hip_bf16.h>

// v2f: A/B operands of V_WMMA_F32_16X16X4_F32 (16x4 f32 = 64 vals / 32 lanes = 2 VGPRs)
typedef __attribute__((ext_vector_type(2))) float v2f;
// v8f: 16x16 f32 C/D accumulator (256 vals / 32 lanes = 8 VGPRs)
typedef __attribute__((ext_vector_type(8))) float v8f;

#define OUT_STRIDE 4100   // 4096 move logits + 4 promo slots per batch row

// One block (256 threads = 8 wave32) per batch element.
//   Phase 1: threads 0..63 tokenize their square and fill LDS from/to score tables.
//   Phase 2: thread 0 writes promo tail (max of from_scores, x4) and values[b]=0.
//   Phase 3: each wave computes two 16x16 tiles of the outer sum with
//            v_wmma_f32_16x16x4_f32 (rank-2 update: [from,1] x [1; to]) and stores.
__global__ void __launch_bounds__(256)
chessformer_fused_kernel(const float* __restrict__ obs,
                         const float* __restrict__ emb,
                         float* __restrict__ out,
                         int B) {
    __shared__ float s_from[64];
    __shared__ float s_to[64];

    const int b   = blockIdx.x;
    const int tid = threadIdx.x;

    const float* obs_b = obs + (size_t)b * 15 * 64;   // (15,8,8) row-major

    // ---- Phase 1: tokenize + embedding gather (threads 0..63, one per square) ----
    if (tid < 64) {
        int token = 0;
        // "first plane > 0.5 wins" == argmax of bool mask; scan high->low so the
        // lowest matching plane index is the final assignment.
        #pragma unroll
        for (int p = 11; p >= 0; --p) {
            float v = obs_b[p * 64 + tid];          // 256B coalesced per plane
            if (v > 0.5f) token = p + 1;
        }
        s_from[tid] = emb[token * 2 + 0];
        s_to[tid]   = emb[token * 2 + 1];
    }
    __syncthreads();

    // ---- Phase 2: promo tail + values (tiny, serial on thread 0) ----
    if (tid == 0) {
        float m = s_from[0];
        #pragma unroll
        for (int i = 1; i < 64; ++i) m = fmaxf(m, s_from[i]);
        float* tail = out + (size_t)b * OUT_STRIDE + 4096;
        tail[0] = m; tail[1] = m; tail[2] = m; tail[3] = m;
        out[(size_t)B * OUT_STRIDE + b] = 0.0f;     // values[b] = 0
    }

    // ---- Phase 3: 16 tiles of 16x16 via WMMA; wave w handles tiles 2w, 2w+1 ----
    // EXEC is all 1s here (no divergence at wave granularity), as WMMA requires.
    const int wave   = tid >> 5;
    const int lane   = tid & 31;
    const int lane16 = lane & 15;
    const bool lo    = lane < 16;                   // lanes 0-15 hold K=0,1; 16-31 hold K=2,3

    float* out_b = out + (size_t)b * OUT_STRIDE;

    #pragma unroll
    for (int i = 0; i < 2; ++i) {
        const int tile  = wave * 2 + i;             // 0..15
        const int fbase = (tile >> 2) * 16;         // from-block
        const int tbase = (tile & 3) * 16;          // to-block

        // A(16x4): col0 = from chunk, col1 = ones, cols 2,3 = zero
        v2f a;
        a.x = lo ? s_from[fbase + lane16] : 0.0f;   // K=0 (lanes<16) / K=2 (zeros)
        a.y = lo ? 1.0f                  : 0.0f;    // K=1             / K=3
        // B(4x16): row0 = ones, row1 = to chunk, rows 2,3 = zero
        v2f bm;
        bm.x = lo ? 1.0f                 : 0.0f;
        bm.y = lo ? s_to[tbase + lane16] : 0.0f;

        v8f acc = {};
        // D[M,N] = from[fbase+M]*1 + 1*to[tbase+N]  (exact f32)
        acc = __builtin_amdgcn_wmma_f32_16x16x4_f32(
                  /*neg_a=*/false, a, /*neg_b=*/false, bm,
                  /*c_mod=*/(short)0, acc, /*reuse_a=*/false, /*reuse_b=*/false);

        const int N = tbase + lane16;               // global "to" index of this lane
        #pragma unroll
        for (int r = 0; r < 8; ++r) {
            const int M = fbase + r + (lo ? 0 : 8); // global "from" index of acc[r]
            float v = acc[r];
            if (M == N) v = 0.0f;                   // zero the from==to diagonal
            out_b[M * 64 + N] = v;                  // lanes 0-15 / 16-31: two 64B rows
        }
    }
}

extern "C" void kernel_launch(void* const* d_in, const int* in_sizes, int n_in,
                              void* d_out, int out_size, void* d_ws, size_t ws_size,
                              hipStream_t stream) {
    (void)n_in; (void)out_size; (void)d_ws; (void)ws_size;
    const float* obs = (const float*)d_in[0];   // (B,15,8,8) f32
    const float* emb = (const float*)d_in[1];   // (13,2) f32
    float* out = (float*)d_out;                 // (B,4100) logits ++ (B,) values, flat

    const int B = in_sizes[0] / (15 * 64);      // 16384 for the reference shapes

    chessformer_fused_kernel<<<dim3(B), dim3(256), 0, stream>>>(obs, emb, out, B);
}